// DNN_Beamformer_30726196036156
// MI455X (gfx1250) — compile-verified
//
#include <hip/hip_runtime.h>
#include <hip/hip_bf16.h>
#include <stdint.h>

// Problem constants (from reference setup_inputs)
constexpr int Bv = 8;
constexpr int Tv = 1000;
constexpr int Cv = 8;
constexpr int Fv = 513;
constexpr int ATTv = 320;
constexpr float EPSV = 1e-15f;

typedef float v2f __attribute__((ext_vector_type(2)));
typedef float v8f __attribute__((ext_vector_type(8)));

// ---------------- scratch layout (float offsets) ----------------
constexpr size_t BFTsz   = (size_t)Bv * Fv * Tv;            // 4,104,000
constexpr size_t OFF_MS  = 0;                                // m_speech (B,F,T)
constexpr size_t OFF_MN  = OFF_MS + BFTsz;                   // m_noise  (B,F,T)
constexpr size_t GSZ     = (size_t)Bv * Fv * 2 * 256;        // Gram mats (B,F,2,16,16)
constexpr size_t OFF_G   = OFF_MN + BFTsz;                   // reduced Grams
constexpr size_t OFF_FEAT= OFF_G + GSZ;                      // feat (B,C,F)
constexpr size_t OFF_E   = OFF_FEAT + (size_t)Bv * Cv * Fv;  // e (B,C) = 64
constexpr size_t OFF_U   = OFF_E + 64;                       // u (B,C) = 64
constexpr size_t OFF_WSV = OFF_U + 64;                       // ws (B,F,C,2)
constexpr int    SEG     = 4;                                // T split for occupancy
constexpr int    TSEG    = Tv / SEG;                         // 250
constexpr size_t OFF_GP  = OFF_WSV + (size_t)Bv * Fv * Cv * 2; // partial Grams (SEG,B,F,2,256)

// =====================================================================
// Kernel 1: per-(b,f) mask mean over channels + normalize over time
// =====================================================================
__global__ __launch_bounds__(256) void mask_kernel(const float* __restrict__ msk_s,
                                                   const float* __restrict__ msk_n,
                                                   float* __restrict__ wsf) {
  __shared__ float lm[Tv];
  __shared__ float red[256];
  const int bf  = blockIdx.x;      // b*F + f
  const int tid = threadIdx.x;
  const size_t mbase = (size_t)bf * Cv * Tv;

  for (int s = 0; s < 2; ++s) {
    const float* mk  = s ? msk_n : msk_s;
    float* outm = wsf + (s ? OFF_MN : OFF_MS) + (size_t)bf * Tv;
    float partial = 0.f;
    for (int t = tid; t < Tv; t += 256) {
      float a = 0.f;
#pragma unroll
      for (int c = 0; c < Cv; ++c) a += mk[mbase + (size_t)c * Tv + t];
      a *= (1.0f / Cv);
      lm[t] = a;
      partial += a;
    }
    red[tid] = partial;
    __syncthreads();
    for (int o = 128; o > 0; o >>= 1) {
      if (tid < o) red[tid] += red[tid + o];
      __syncthreads();
    }
    const float scale = 1.f / (red[0] + EPSV);
    __syncthreads();
    for (int t = tid; t < Tv; t += 256) outm[t] = lm[t] * scale;
    __syncthreads();
  }
}

// =====================================================================
// Kernel 2: partial PSD Gram matrices via V_WMMA_F32_16X16X4_F32
//   One wave per frequency; 16 freqs / 512-thread block.
//   T split into SEG segments across blocks for occupancy; each block
//   stages its x tile into LDS with global_load_async_to_lds_b32.
//   G = sum_t m_t * u(t) u(t)^T,  u = [xr0..xr7, xi0..xi7]
// =====================================================================
#define TCHUNK 48
#define NCHSEG 6    // ceil(250/48)
#define LSTR   49   // padded LDS stride (conflict-free)

__global__ __launch_bounds__(512) void psd_kernel(const float* __restrict__ dre,
                                                  const float* __restrict__ dim,
                                                  float* __restrict__ wsf) {
  __shared__ float lx[16 * 16 * LSTR];   // [f_local][comp(16)][t]
  __shared__ float lmm[2 * 16 * LSTR];   // [mask][f_local][t]

  const int tid  = threadIdx.x;
  const int wave = tid >> 5;
  const int lane = tid & 31;

  const int seg  = blockIdx.x / (Bv * 33);
  const int rem  = blockIdx.x % (Bv * 33);
  const int bb   = rem / 33;
  const int f0   = (rem % 33) * 16;
  const int f    = f0 + wave;            // this wave's frequency
  const int tstart = seg * TSEG;
  const int tend   = tstart + TSEG;

  const int comp  = lane & 15;           // A row / B col component index
  const int khalf = (lane >> 4) << 1;    // lanes 16-31 hold K=2,3

  v8f accS[2] = {{}, {}};                // split accumulation chains (k parity)
  v8f accN[2] = {{}, {}};

  for (int ch = 0; ch < NCHSEG; ++ch) {
    const int t0 = tstart + ch * TCHUNK;

    // ---- async staging of x tile: (48 t) x (8 c) x (16 f) ----
    for (int i = tid; i < TCHUNK * 8 * 16; i += 512) {
      const int fl = i & 15;
      const int c  = (i >> 4) & 7;
      const int tl = i >> 7;
      const int t  = t0 + tl;
      const int fg = f0 + fl;
      const int idxR = (fl * 16 + c) * LSTR + tl;
      const int idxI = (fl * 16 + c + 8) * LSTR + tl;
      if (t < tend && fg < Fv) {
        const size_t gi = ((size_t)(bb * Tv + t) * Cv + c) * Fv + fg;
        const uint32_t ldsR = (uint32_t)(size_t)&lx[idxR];
        const uint32_t ldsI = (uint32_t)(size_t)&lx[idxI];
        const float* gpr = dre + gi;
        const float* gpi = dim + gi;
        // CDNA5 async Global->LDS copy (ASYNCcnt-tracked)
        asm volatile("global_load_async_to_lds_b32 %0, %1, off"
                     :: "v"(ldsR), "v"(gpr) : "memory");
        asm volatile("global_load_async_to_lds_b32 %0, %1, off"
                     :: "v"(ldsI), "v"(gpi) : "memory");
        // prefetch next chunk (one per 64B run)
        if ((i & 15) == 0 && t + TCHUNK < tend) {
          __builtin_prefetch(gpr + (size_t)TCHUNK * Cv * Fv, 0, 0);
          __builtin_prefetch(gpi + (size_t)TCHUNK * Cv * Fv, 0, 0);
        }
      } else {
        lx[idxR] = 0.f;
        lx[idxI] = 0.f;
      }
    }
    // ---- cooperative load of normalized masks (regular path) ----
    for (int i = tid; i < 2 * 16 * TCHUNK; i += 512) {
      const int tl = i % TCHUNK;
      const int fl = (i / TCHUNK) & 15;
      const int s  = i / (16 * TCHUNK);
      const int t  = t0 + tl;
      const int fg = f0 + fl;
      float v = 0.f;
      if (t < tend && fg < Fv) {
        const float* mp = wsf + (s ? OFF_MN : OFF_MS);
        v = mp[(size_t)(bb * Fv + fg) * Tv + t];
      }
      lmm[(s * 16 + fl) * LSTR + tl] = v;
    }
    // compiler cannot see ASYNCcnt: drain it before the barrier
    asm volatile("s_wait_asynccnt 0x0" ::: "memory");
    __syncthreads();

    if (f < Fv) {                         // wave-uniform -> EXEC stays all-ones
      const float* xr = &lx[(wave * 16 + comp) * LSTR];
      const float* m0 = &lmm[(0 * 16 + wave) * LSTR];
      const float* m1 = &lmm[(1 * 16 + wave) * LSTR];
#pragma unroll
      for (int kk = 0; kk < TCHUNK / 4; ++kk) {
        const int ta = kk * 4 + khalf;
        const float x0 = xr[ta], x1 = xr[ta + 1];
        const float s0 = m0[ta], s1 = m0[ta + 1];
        const float n0 = m1[ta], n1 = m1[ta + 1];
        v2f bv; bv.x = x0;      bv.y = x1;        // B: raw components
        v2f as; as.x = s0 * x0; as.y = s1 * x1;   // A: speech-mask-weighted
        v2f an; an.x = n0 * x0; an.y = n1 * x1;   // A: noise-mask-weighted
        accS[kk & 1] = __builtin_amdgcn_wmma_f32_16x16x4_f32(
            false, as, false, bv, (short)0, accS[kk & 1], false, false);
        accN[kk & 1] = __builtin_amdgcn_wmma_f32_16x16x4_f32(
            false, an, false, bv, (short)0, accN[kk & 1], false, false);
      }
    }
    __syncthreads();
  }

  // ---- write partial 16x16 Gram matrices (speech then noise) ----
  if (f < Fv) {
    const v8f gs = accS[0] + accS[1];
    const v8f gn = accN[0] + accN[1];
    float* g = wsf + OFF_GP + ((size_t)seg * (Bv * Fv) + (size_t)(bb * Fv + f)) * 512;
    const int M0 = (lane >> 4) * 8;
    const int N  = lane & 15;
#pragma unroll
    for (int r = 0; r < 8; ++r) {
      g[(M0 + r) * 16 + N]       = gs[r];
      g[256 + (M0 + r) * 16 + N] = gn[r];
    }
  }
}

// =====================================================================
// Kernel 2b: reduce partial Grams over segments
// =====================================================================
__global__ __launch_bounds__(256) void psd_reduce_kernel(float* __restrict__ wsf) {
  const size_t bf = blockIdx.x;
  for (int i = threadIdx.x; i < 512; i += 256) {
    float s = 0.f;
#pragma unroll
    for (int seg = 0; seg < SEG; ++seg)
      s += wsf[OFF_GP + ((size_t)seg * (Bv * Fv) + bf) * 512 + i];
    wsf[OFF_G + bf * 512 + i] = s;
  }
}

// =====================================================================
// Kernel 3: feat[b,c,f] = | sum_{e!=c} psd_s[c,e] / (C-1) |
// =====================================================================
__global__ void feat_kernel(float* __restrict__ wsf) {
  const int bf = blockIdx.x;
  const int bb = bf / Fv;
  const int ff = bf % Fv;
  const int c  = threadIdx.x;
  if (c >= Cv) return;
  const float* g = wsf + OFF_G + (size_t)bf * 512;   // speech Gram
  float pr = 0.f, pi = 0.f;
#pragma unroll
  for (int e = 0; e < 8; ++e) {
    if (e == c) continue;
    pr += g[c * 16 + e] + g[(c + 8) * 16 + e + 8];
    pi += g[(c + 8) * 16 + e] - g[c * 16 + e + 8];
  }
  pr *= (1.f / 7.f);
  pi *= (1.f / 7.f);
  wsf[OFF_FEAT + ((size_t)bb * Cv + c) * Fv + ff] = sqrtf(pr * pr + pi * pi);
}

// =====================================================================
// Kernel 4: attention MLP  e[b,c] = tanh(feat @ W^T + b) @ w_g + b_g
// =====================================================================
__global__ __launch_bounds__(ATTv) void att_kernel(const float* __restrict__ Wp,
                                                   const float* __restrict__ bp,
                                                   const float* __restrict__ wg,
                                                   const float* __restrict__ bg,
                                                   float* __restrict__ wsf) {
  __shared__ float red[ATTv];
  const int bc = blockIdx.x;
  const int j  = threadIdx.x;
  const float* feat = wsf + OFF_FEAT + (size_t)bc * Fv;
  const float* wrow = Wp + (size_t)j * Fv;
  float acc = bp[j];
  for (int fq = 0; fq < Fv; ++fq) acc += feat[fq] * wrow[fq];
  red[j] = tanhf(acc) * wg[j];
  __syncthreads();
  if (j == 0) {
    float s = bg[0];
    for (int k = 0; k < ATTv; ++k) s += red[k];
    wsf[OFF_E + bc] = s;
  }
}

// =====================================================================
// Kernel 5: u = softmax(2*e) over channels, per batch
// =====================================================================
__global__ void softmax_kernel(float* __restrict__ wsf) {
  const int tid = threadIdx.x;     // 64 threads = B*C
  const int bb  = tid >> 3;
  __shared__ float ev[64];
  ev[tid] = 2.f * wsf[OFF_E + tid];
  __syncthreads();
  float mx = -1e30f;
  for (int k = 0; k < 8; ++k) mx = fmaxf(mx, ev[bb * 8 + k]);
  const float ex = __expf(ev[tid] - mx);
  __syncthreads();
  ev[tid] = ex;
  __syncthreads();
  float s = 0.f;
  for (int k = 0; k < 8; ++k) s += ev[bb * 8 + k];
  wsf[OFF_U + tid] = ex / s;
}

// =====================================================================
// Kernel 6: per-(b,f) MVDR: num = psd_n^{-1} psd_s (Gauss-Jordan, one
// wave per (b,f), lane = column of augmented [psd_n | psd_s]),
// ws = (num / (tr(num)+eps)) @ u
// =====================================================================
__global__ __launch_bounds__(128) void solve_kernel(float* __restrict__ wsf) {
  const int wave = threadIdx.x >> 5;
  const int lane = threadIdx.x & 31;
  const int idx  = blockIdx.x * 4 + wave;
  if (idx >= Bv * Fv) return;
  const int bb = idx / Fv;

  const float* gs = wsf + OFF_G + (size_t)idx * 512;  // speech Gram
  const float* gn = gs + 256;                         // noise Gram

  // lane j (<16) owns column j of the augmented complex system [N | S]
  float ar[8], ai[8];
  const float* g  = (lane < 8) ? gn : gs;
  const int    cN = lane & 7;
#pragma unroll
  for (int r = 0; r < 8; ++r) {
    if (lane < 16) {
      ar[r] = g[r * 16 + cN] + g[(r + 8) * 16 + cN + 8];       // Re
      ai[r] = g[(r + 8) * 16 + cN] - g[r * 16 + cN + 8];       // Im
    } else {
      ar[r] = 0.f; ai[r] = 0.f;
    }
  }

  // Gauss-Jordan (no pivoting; psd_n is PSD-dominant)
#pragma unroll
  for (int k = 0; k < 8; ++k) {
    const float pr  = __shfl(ar[k], k, 32);
    const float pi  = __shfl(ai[k], k, 32);
    const float inv = 1.f / (pr * pr + pi * pi + 1e-30f);
    const float nr  = (ar[k] * pr + ai[k] * pi) * inv;   // row k /= pivot
    const float ni  = (ai[k] * pr - ar[k] * pi) * inv;
    ar[k] = nr; ai[k] = ni;
#pragma unroll
    for (int r = 0; r < 8; ++r) {
      if (r == k) continue;
      const float fr = __shfl(ar[r], k, 32);   // A[r][k] from lane k
      const float fi = __shfl(ai[r], k, 32);
      const float dr = fr * ar[k] - fi * ai[k];
      const float di = fr * ai[k] + fi * ar[k];
      ar[r] -= dr;
      ai[r] -= di;
    }
  }
  // lanes 8..15 now hold num = psd_n^{-1} psd_s columns.

  // trace(num)
  float tr_r = 0.f, tr_i = 0.f;
#pragma unroll
  for (int r = 0; r < 8; ++r) {
    if (lane == 8 + r) { tr_r = ar[r]; tr_i = ai[r]; }
  }
#pragma unroll
  for (int off = 16; off > 0; off >>= 1) {
    tr_r += __shfl_xor(tr_r, off, 32);
    tr_i += __shfl_xor(tr_i, off, 32);
  }
  tr_r += EPSV;
  const float den = 1.f / (tr_r * tr_r + tr_i * tr_i);

  float uc = 0.f;
  if (lane >= 8 && lane < 16) uc = wsf[OFF_U + bb * 8 + (lane - 8)];

  float* wsp = wsf + OFF_WSV + (size_t)idx * 16;
#pragma unroll
  for (int e = 0; e < 8; ++e) {
    float vr = ar[e] * uc;     // lanes outside 8..15 contribute 0
    float vi = ai[e] * uc;
#pragma unroll
    for (int off = 16; off > 0; off >>= 1) {
      vr += __shfl_xor(vr, off, 32);
      vi += __shfl_xor(vi, off, 32);
    }
    if (lane == 0) {
      wsp[e * 2 + 0] = (vr * tr_r + vi * tr_i) * den;   // (num@u)/tr
      wsp[e * 2 + 1] = (vi * tr_r - vr * tr_i) * den;
    }
  }
}

// =====================================================================
// Kernel 7: enh[b,t,f] = sum_c conj(ws[b,f,c]) * x[b,t,c,f]
// out = [ real(B,T,F) | imag(B,T,F) ]
// =====================================================================
__global__ __launch_bounds__(256) void beam_kernel(const float* __restrict__ dre,
                                                   const float* __restrict__ dim,
                                                   const float* __restrict__ wsf,
                                                   float* __restrict__ out) {
  const int bt = blockIdx.x;
  const int bb = bt / Tv;
  const size_t xbase = (size_t)bt * Cv * Fv;
  const size_t obase = (size_t)bt * Fv;
  const size_t imoff = (size_t)Bv * Tv * Fv;
  for (int f = threadIdx.x; f < Fv; f += 256) {
    const float* wp = wsf + OFF_WSV + (size_t)(bb * Fv + f) * 16;
    float accr = 0.f, acci = 0.f;
#pragma unroll
    for (int c = 0; c < Cv; ++c) {
      const float wr = wp[c * 2];
      const float wi = wp[c * 2 + 1];
      const float xr = dre[xbase + (size_t)c * Fv + f];
      const float xi = dim[xbase + (size_t)c * Fv + f];
      accr += wr * xr + wi * xi;     // conj(w) * x
      acci += wr * xi - wi * xr;
    }
    out[obase + f]         = accr;
    out[imoff + obase + f] = acci;
  }
}

// =====================================================================
extern "C" void kernel_launch(void* const* d_in, const int* in_sizes, int n_in,
                              void* d_out, int out_size, void* d_ws, size_t ws_size,
                              hipStream_t stream) {
  (void)in_sizes; (void)n_in; (void)out_size; (void)ws_size;
  const float* dre   = (const float*)d_in[0];   // data_real  (B,T,C,F)
  const float* dimg  = (const float*)d_in[1];   // data_imag  (B,T,C,F)
  const float* msk_s = (const float*)d_in[2];   // mask_speech (B,F,C,T)
  const float* msk_n = (const float*)d_in[3];   // mask_noise  (B,F,C,T)
  const float* Wp    = (const float*)d_in[4];   // (att,F)
  const float* bp    = (const float*)d_in[5];   // (att,)
  const float* wg    = (const float*)d_in[6];   // (att,)
  const float* bg    = (const float*)d_in[7];   // (1,)
  float* wsf = (float*)d_ws;
  float* out = (float*)d_out;

  hipLaunchKernelGGL(mask_kernel,      dim3(Bv * Fv),           dim3(256), 0, stream, msk_s, msk_n, wsf);
  hipLaunchKernelGGL(psd_kernel,       dim3(SEG * Bv * 33),     dim3(512), 0, stream, dre, dimg, wsf);
  hipLaunchKernelGGL(psd_reduce_kernel,dim3(Bv * Fv),           dim3(256), 0, stream, wsf);
  hipLaunchKernelGGL(feat_kernel,      dim3(Bv * Fv),           dim3(8),   0, stream, wsf);
  hipLaunchKernelGGL(att_kernel,       dim3(Bv * Cv),           dim3(ATTv),0, stream, Wp, bp, wg, bg, wsf);
  hipLaunchKernelGGL(softmax_kernel,   dim3(1),                 dim3(64),  0, stream, wsf);
  hipLaunchKernelGGL(solve_kernel,     dim3((Bv * Fv + 3) / 4), dim3(128), 0, stream, wsf);
  hipLaunchKernelGGL(beam_kernel,      dim3(Bv * Tv),           dim3(256), 0, stream, dre, dimg, wsf, out);
}